// LightGCN_17111149707373
// MI455X (gfx1250) — compile-verified
//
#include <hip/hip_runtime.h>
#include <hip/hip_bf16.h>
#include <stdint.h>

// LightGCN propagation on MI455X (gfx1250, wave32).
// Pipeline: degree histogram -> exclusive scan -> CSC bucket fill (with
// precomputed symmetric-norm edge weights) -> 3x atomic-free gather SpMM.
// Per-block edge metadata is staged into LDS with the Tensor Data Mover
// (TENSOR_LOAD_TO_LDS + s_wait_tensorcnt), then broadcast-read from LDS with
// a software-pipelined single ds_load_b64 per edge.

#define N_FEAT     64
#define SCAN_BS    256
#define NUM_LAYERS 3
#define WAVES_PER_BLOCK 8
#define TDM_CAP    1024   // LDS metadata tile capacity (8 KB); avg span ~256

#if defined(__HIP_DEVICE_COMPILE__) && __has_builtin(__builtin_amdgcn_tensor_load_to_lds) && __has_builtin(__builtin_amdgcn_s_wait_tensorcnt)
#define HAS_TDM 1
#else
#define HAS_TDM 0
#endif

#if HAS_TDM
typedef uint32_t v4u __attribute__((ext_vector_type(4)));
typedef int32_t  v8i __attribute__((ext_vector_type(8)));
typedef int32_t  v4i __attribute__((ext_vector_type(4)));

// Issue a 1-D TDM load of `len` 8-byte elements: global `gaddr` -> LDS offset
// `lds_addr`. D# packing per CDNA5 ISA 8.3-8.6 (count=1, data_size=3 (8B),
// tensor_dim0 = tile_dim0 = len, tensor_dim1 = 1, type=2, no pad/iterate).
__device__ __forceinline__ void tdm_load_1d(uint32_t lds_addr, uint64_t gaddr,
                                            int len) {
    v4u g0 = { 1u,                                   // count=1, user mode
               lds_addr,                             // D#.lds_addr
               (uint32_t)gaddr,                      // global_addr[31:0]
               (uint32_t)((gaddr >> 32) & 0x01FFFFFFu) | (2u << 30) }; // +type=2
    v8i g1 = { (int)(3u << 16),                      // data_size=3 (8B), mask=0
               (int)((len & 0xFFFF) << 16),          // tensor_dim0[15:0]
               (int)(((len >> 16) & 0xFFFF) | (1u << 16)), // dim0[31:16], dim1=1
               (int)((len & 0xFFFF) << 16),          // tile_dim0 = len
               0,                                    // tile_dim1=0, tile_dim2=0
               len,                                  // tensor_dim0_stride lo
               0, 0 };
    v4i gz = { 0, 0, 0, 0 };
#if __clang_major__ >= 23
    v8i gz8 = { 0, 0, 0, 0, 0, 0, 0, 0 };
    __builtin_amdgcn_tensor_load_to_lds(g0, g1, gz, gz, gz8, 0);
#else
    __builtin_amdgcn_tensor_load_to_lds(g0, g1, gz, gz, 0);
#endif
}
#endif // HAS_TDM

// ---------------------------------------------------------------- init ----
__global__ void zero_kernel(unsigned int* __restrict__ deg,
                            unsigned int* __restrict__ cursor,
                            float* __restrict__ out,
                            int n_nodes, int out_elems) {
    int i = blockIdx.x * blockDim.x + threadIdx.x;
    int stride = gridDim.x * blockDim.x;
    for (int k = i; k < out_elems; k += stride) out[k] = 0.0f;
    for (int k = i; k < n_nodes; k += stride) { deg[k] = 0u; cursor[k] = 0u; }
}

// ------------------------------------------------------- degree histogram --
__global__ void degree_kernel(const long long* __restrict__ ei,
                              unsigned int* __restrict__ deg, int E) {
    int i = blockIdx.x * blockDim.x + threadIdx.x;
    int stride = gridDim.x * blockDim.x;
    for (int e = i; e < E; e += stride) {
        long long c = __builtin_nontemporal_load(&ei[(long long)E + e]);
        __builtin_prefetch(&ei[(long long)E + e + stride], 0, 0);
        atomicAdd(&deg[(int)c], 1u);
    }
}

// ----------------------------------------------------------- deg^{-1/2} ----
__global__ void dinv_kernel(const unsigned int* __restrict__ deg,
                            float* __restrict__ dinv, int n) {
    int i = blockIdx.x * blockDim.x + threadIdx.x;
    if (i < n) {
        unsigned int d = deg[i];
        dinv[i] = (d > 0u) ? rsqrtf((float)d) : 0.0f;
    }
}

// ------------------------------------------------- exclusive scan (3-pass) --
__global__ void scan_block_kernel(const unsigned int* __restrict__ deg,
                                  int* __restrict__ offsets,
                                  int* __restrict__ blocksums, int n) {
    __shared__ int sm[SCAN_BS];
    int t = threadIdx.x;
    int i = blockIdx.x * SCAN_BS + t;
    int v = (i < n) ? (int)deg[i] : 0;
    sm[t] = v;
    __syncthreads();
    for (int off = 1; off < SCAN_BS; off <<= 1) {
        int val = sm[t];
        if (t >= off) val += sm[t - off];
        __syncthreads();
        sm[t] = val;
        __syncthreads();
    }
    int incl = sm[t];
    if (i < n) offsets[i] = incl - v;
    if (t == SCAN_BS - 1) blocksums[blockIdx.x] = incl;
}

__global__ void scan_sums_kernel(int* __restrict__ blocksums, int nblocks) {
    __shared__ int sm[1024];
    int t = threadIdx.x;
    int v = (t < nblocks) ? blocksums[t] : 0;
    sm[t] = v;
    __syncthreads();
    for (int off = 1; off < 1024; off <<= 1) {
        int val = sm[t];
        if (t >= off) val += sm[t - off];
        __syncthreads();
        sm[t] = val;
        __syncthreads();
    }
    if (t < nblocks) blocksums[t] = sm[t] - v;
}

__global__ void scan_add_kernel(int* __restrict__ offsets,
                                const int* __restrict__ blocksums, int n) {
    int i = blockIdx.x * blockDim.x + threadIdx.x;
    if (i < n) offsets[i] += blocksums[i / SCAN_BS];
}

// ----------------------------------------------------------- CSC fill ------
__global__ void fill_kernel(const long long* __restrict__ ei,
                            const int* __restrict__ offsets,
                            unsigned int* __restrict__ cursor,
                            const float* __restrict__ dinv,
                            int2* __restrict__ nbr, int E) {
    int i = blockIdx.x * blockDim.x + threadIdx.x;
    int stride = gridDim.x * blockDim.x;
    for (int e = i; e < E; e += stride) {
        long long s64 = __builtin_nontemporal_load(&ei[e]);
        long long d64 = __builtin_nontemporal_load(&ei[(long long)E + e]);
        __builtin_prefetch(&ei[e + stride], 0, 0);
        int s = (int)s64, d = (int)d64;
        unsigned int p = atomicAdd(&cursor[d], 1u);
        int slot = offsets[d] + (int)p;
        int2 pk;
        pk.x = s;
        pk.y = __float_as_int(dinv[s] * dinv[d]);
        nbr[slot] = pk;
    }
}

// ------------------------------------------------ gather SpMM per layer ----
// One wave32 per destination node; lane l owns features [2l, 2l+1].
// Block's contiguous CSC span is DMA'd to LDS by the TDM once; the edge loop
// is software-pipelined: one ds_load_b64 per edge, with the next feature row
// prefetched (global_prefetch) while the current row is FMA'd.
__global__ void propagate_kernel(const float* __restrict__ src,
                                 float* __restrict__ dst,
                                 float* __restrict__ out,
                                 const int2* __restrict__ nbr,
                                 const int* __restrict__ offsets,
                                 const unsigned int* __restrict__ deg,
                                 float alpha, int n_nodes) {
    __shared__ int2 smem[TDM_CAP];

    const int lane = threadIdx.x & 31;
    const int v0   = blockIdx.x * WAVES_PER_BLOCK;
    const int v    = v0 + (int)(threadIdx.x >> 5);

    // Contiguous metadata span covered by this block's nodes
    int vLast = v0 + WAVES_PER_BLOCK - 1;
    if (vLast >= n_nodes) vLast = n_nodes - 1;
    const int eBeg = offsets[v0];
    const int eEnd = offsets[vLast] + (int)deg[vLast];
    const int len  = eEnd - eBeg;

    bool useLds = false;
#if HAS_TDM
    useLds = (len > 0) && (len <= TDM_CAP);
    if (useLds && threadIdx.x == 0) {
        tdm_load_1d((uint32_t)(uintptr_t)&smem[0],
                    (uint64_t)(uintptr_t)(nbr + eBeg), len);
        __builtin_amdgcn_s_wait_tensorcnt(0);      // s_wait_tensorcnt 0
    }
    __syncthreads();
#endif

    if (v >= n_nodes) return;
    const int beg = offsets[v];
    const int cnt = (int)deg[v];

    float2 acc = make_float2(0.0f, 0.0f);
    if (useLds) {
        const int lbase = beg - eBeg;
        int2 pk = (cnt > 0) ? smem[lbase] : make_int2(0, 0);
        for (int j = 0; j < cnt; ++j) {
            // stage j+1 metadata (clamped) while using j's
            int jn = j + 1;
            if (jn >= cnt) jn = cnt - 1;
            int2 nxt = smem[lbase + jn];           // single ds_load_b64 / edge
            __builtin_prefetch(src + (size_t)nxt.x * N_FEAT, 0, 0);
            const float2 hv =
                *((const float2*)(src + (size_t)pk.x * N_FEAT) + lane);
            const float wj = __int_as_float(pk.y);
            acc.x += wj * hv.x;
            acc.y += wj * hv.y;
            pk = nxt;
        }
    } else {
        // Fallback: coalesced global metadata loads + wave32 shuffle broadcast
        for (int base = 0; base < cnt; base += 32) {
            int e = base + lane;
            int s = 0;
            float w = 0.0f;
            if (e < cnt) {
                int2 pk = nbr[beg + e];
                s = pk.x;
                w = __int_as_float(pk.y);
            }
            __builtin_prefetch(&nbr[beg + base + 32], 0, 0);
            int m = cnt - base;
            if (m > 32) m = 32;
            for (int j = 0; j < m; ++j) {
                int   sj = __shfl(s, j, 32);
                float wj = __shfl(w, j, 32);
                const float2 hv =
                    *((const float2*)(src + (size_t)sj * N_FEAT) + lane);
                acc.x += wj * hv.x;
                acc.y += wj * hv.y;
            }
        }
    }

    size_t o = (size_t)v * N_FEAT + (size_t)lane * 2;
    *(float2*)(dst + o) = acc;                    // feeds next layer
    float2 ov = *(float2*)(out + o);              // out += alpha * h
    ov.x += alpha * acc.x;
    ov.y += alpha * acc.y;
    *(float2*)(out + o) = ov;
}

// ---------------------------------------------------------------- launch ---
static inline size_t align256(size_t v) { return (v + 255) & ~(size_t)255; }

extern "C" void kernel_launch(void* const* d_in, const int* in_sizes, int n_in,
                              void* d_out, int out_size, void* d_ws, size_t ws_size,
                              hipStream_t stream) {
    const float*     x  = (const float*)d_in[0];
    const long long* ei = (const long long*)d_in[1];

    const int n_nodes = in_sizes[0] / N_FEAT;     // 150000
    const int E       = in_sizes[1] / 2;          // 4800000
    float* out = (float*)d_out;

    // Workspace carve-up (~118 MB total)
    char* p = (char*)d_ws;
    unsigned int* deg    = (unsigned int*)p; p += align256((size_t)n_nodes * 4);
    unsigned int* cursor = (unsigned int*)p; p += align256((size_t)n_nodes * 4);
    float*        dinv   = (float*)p;        p += align256((size_t)n_nodes * 4);
    int*          offs   = (int*)p;          p += align256((size_t)n_nodes * 4);
    int*          bsums  = (int*)p;          p += align256((size_t)1024 * 4);
    int2*         nbr    = (int2*)p;         p += align256((size_t)E * 8);
    float*        hA     = (float*)p;        p += align256((size_t)n_nodes * N_FEAT * 4);
    float*        hB     = (float*)p;

    const int BS = 256;
    const int out_elems = n_nodes * N_FEAT;

    zero_kernel<<<2048, BS, 0, stream>>>(deg, cursor, out, n_nodes, out_elems);
    degree_kernel<<<4096, BS, 0, stream>>>(ei, deg, E);
    dinv_kernel<<<(n_nodes + BS - 1) / BS, BS, 0, stream>>>(deg, dinv, n_nodes);

    const int nScanBlocks = (n_nodes + SCAN_BS - 1) / SCAN_BS;   // 586 <= 1024
    scan_block_kernel<<<nScanBlocks, SCAN_BS, 0, stream>>>(deg, offs, bsums, n_nodes);
    scan_sums_kernel<<<1, 1024, 0, stream>>>(bsums, nScanBlocks);
    scan_add_kernel<<<nScanBlocks, SCAN_BS, 0, stream>>>(offs, bsums, n_nodes);

    fill_kernel<<<4096, BS, 0, stream>>>(ei, offs, cursor, dinv, nbr, E);

    const float alpha = 1.0f / (1.0f + (float)NUM_LAYERS);
    const int propBlocks = (n_nodes + WAVES_PER_BLOCK - 1) / WAVES_PER_BLOCK;
    propagate_kernel<<<propBlocks, BS, 0, stream>>>(x,  hA, out, nbr, offs, deg, alpha, n_nodes);
    propagate_kernel<<<propBlocks, BS, 0, stream>>>(hA, hB, out, nbr, offs, deg, alpha, n_nodes);
    propagate_kernel<<<propBlocks, BS, 0, stream>>>(hB, hA, out, nbr, offs, deg, alpha, n_nodes);
}